// TorchPQIndexBase_53506702574184
// MI455X (gfx1250) — compile-verified
//
#include <hip/hip_runtime.h>
#include <stdint.h>

typedef __attribute__((ext_vector_type(2))) float v2f;
typedef __attribute__((ext_vector_type(8))) float v8f;
typedef __attribute__((ext_vector_type(4))) unsigned int u32x4;
typedef __attribute__((ext_vector_type(8))) int i32x8;
typedef __attribute__((ext_vector_type(4))) int i32x4;

#define BATCH    32
#define M_SUB    64
#define KSUB_    256
#define DSUB_    4
#define TOPK     64
#define BQ       4            // queries per scan block (interleaved table, 256KB LDS)
#define NSLICE   64           // slices per query for select pass 1
#define SLICE_CAP 15625       // 1e6 / 64
#define FLT_INF  __builtin_inff()

#if defined(__has_builtin)
# if __has_builtin(__builtin_amdgcn_tensor_load_to_lds) && \
     __has_builtin(__builtin_amdgcn_s_wait_tensorcnt)
#  define HAVE_TDM 1
# endif
#endif
#ifndef HAVE_TDM
# define HAVE_TDM 0
#endif

// ---------------------------------------------------------------------------
// Kernel 1: PQ distance table via V_WMMA_F32_16X16X4_F32.
// table[b,m,k] = ||q||^2 + ||c||^2 - 2 q.c ; the -2 q.c term is WMMA, the
// norms are folded into the C accumulator operand. Output layout interleaved
// for the scan: tableI[b>>2][m][k][b&3]  (4-query groups contiguous).
// ---------------------------------------------------------------------------
__global__ __launch_bounds__(32)
void pq_table_wmma(const float* __restrict__ querys,
                   const float* __restrict__ codewords,
                   float* __restrict__ tableI) {
  const int m     = blockIdx.x;    // subspace 0..63
  const int btile = blockIdx.y;    // query tile 0..1
  const int lane  = threadIdx.x;   // 0..31
  const int row   = lane & 15;
  const int kh    = lane >> 4;     // half-wave selects K (dsub) pair {0,1}/{2,3}
  const int bq    = btile * 16 + row;

  // A operand (16x4 f32): A[i,d] = -2 * q[b0+i, m, d]
  const float* qp = querys + (size_t)bq * (M_SUB * DSUB_) + m * DSUB_;
  v2f a;
  a.x = -2.0f * qp[2 * kh + 0];
  a.y = -2.0f * qp[2 * kh + 1];

  // ||q||^2 for this row: (-2q)^2/4 over my 2 dims + partner half's 2 dims
  float pq = 0.25f * (a.x * a.x + a.y * a.y);
  float qn = pq + __shfl_xor(pq, 16, 32);   // lane L holds qnorm of row (L&15)

  const float* cw = codewords + (size_t)m * (KSUB_ * DSUB_);

  for (int kt = 0; kt < 16; ++kt) {
    const int col = row;                    // B/C/D column = lane & 15
    const int ks  = kt * 16 + col;

    // B operand (4x16 f32): B[d,j] = c[m, j, d]; half-wave selects d pair
    v2f b;
    b.x = cw[(size_t)ks * DSUB_ + 2 * kh + 0];
    b.y = cw[(size_t)ks * DSUB_ + 2 * kh + 1];
    float pc = b.x * b.x + b.y * b.y;
    float cn = pc + __shfl_xor(pc, 16, 32); // ||c_mk||^2 for my column

    // C[i,j] = qnorm[i] + cnorm[j]; C layout: VGPR r, this lane -> row r+8*kh
    v8f c;
#pragma unroll
    for (int r = 0; r < 8; ++r) {
      float qn_r = __shfl(qn, r + 8 * kh, 32);
      c[r] = qn_r + cn;
    }

    // D = (-2Q) x C^T + (qn+cn)  ->  squared-L2 table tile
    v8f d = __builtin_amdgcn_wmma_f32_16x16x4_f32(
        false, a, false, b, (short)0, c, false, false);

#pragma unroll
    for (int r = 0; r < 8; ++r) {
      int brow = btile * 16 + r + 8 * kh;
      size_t o = (((size_t)(brow >> 2) * M_SUB + m) * KSUB_ + kt * 16 + col) * BQ
                 + (brow & 3);
      tableI[o] = d[r];
    }
  }
}

// ---------------------------------------------------------------------------
// Kernel 2: pack int32 codes (N,64) -> u8 (N,64): 256MB -> 64MB of traffic
// for every later pass over the codes.
// ---------------------------------------------------------------------------
__global__ void pq_pack_codes(const int* __restrict__ codes,
                              uint32_t* __restrict__ packed, int N) {
  int n = blockIdx.x * blockDim.x + threadIdx.x;
  if (n >= N) return;
  const int4* src = (const int4*)(codes + (size_t)n * 64);
  uint32_t* dst = packed + (size_t)n * 16;
#pragma unroll
  for (int w = 0; w < 16; ++w) {
    int4 v = src[w];
    dst[w] = (uint32_t)(v.x & 255) | ((uint32_t)(v.y & 255) << 8) |
             ((uint32_t)(v.z & 255) << 16) | ((uint32_t)(v.w & 255) << 24);
  }
}

// ---------------------------------------------------------------------------
// Kernel 3: ADC scan. Interleaved table chunk for a 4-query group staged into
// 256KB LDS by the Tensor Data Mover (one descriptor, one wave, TENSORcnt).
// Each (m, code) lookup is a single ds_load_b128 returning all 4 queries.
// ---------------------------------------------------------------------------
__global__ __launch_bounds__(256)
void pq_scan(const uint32_t* __restrict__ packed,
             const float* __restrict__ tableI,
             float* __restrict__ dists, int N) {
  extern __shared__ float tbl[];           // [64][256][4] floats = 256KB
  const int qg  = blockIdx.y;              // 4-query group
  const int b0  = qg * BQ;
  const int tid = threadIdx.x;
  const float* src = tableI + (size_t)qg * (M_SUB * KSUB_ * BQ);

#if HAVE_TDM
  if (tid < 32) {                          // one wave issues the DMA
    uint64_t ga = (uint64_t)(uintptr_t)src;
    // D# group 0: count=1, lds_addr=0, global_addr, type=2 ("image")
    u32x4 g0;
    g0.x = 1u;
    g0.y = 0u;                                   // tbl[] sits at LDS base
    g0.z = (uint32_t)ga;                         // global_addr[31:0]
    g0.w = (uint32_t)((ga >> 32) & 0x01FFFFFFu) | (2u << 30);
    // D# group 1: 2D copy, data_size=4B, 16384 elems x 4 rows, stride 16384
    i32x8 g1;
    g1[0] = 0x20000;            // data_size=2 (4B); workgroup_mask=0
    g1[1] = 0x4000 << 16;       // tensor_dim0[15:0] = 16384 (bits 63:48)
    g1[2] = 4 << 16;            // tensor_dim1[15:0] = 4     (bits 95:80)
    g1[3] = 0x4000 << 16;       // tile_dim0 = 16384         (bits 127:112)
    g1[4] = 4;                  // tile_dim1 = 4             (bits 143:128)
    g1[5] = 16384;              // tensor_dim0_stride[31:0]
    g1[6] = 0;
    g1[7] = 0;
    i32x4 z4 = {0, 0, 0, 0};
#if __clang_major__ >= 23
    i32x8 z8 = {0, 0, 0, 0, 0, 0, 0, 0};
    __builtin_amdgcn_tensor_load_to_lds(g0, g1, z4, z4, z8, 0);
#else
    __builtin_amdgcn_tensor_load_to_lds(g0, g1, z4, z4, 0);
#endif
    __builtin_amdgcn_s_wait_tensorcnt(0);
  }
  __syncthreads();
#else
  {
    const float4* s4 = (const float4*)src;
    float4* d4 = (float4*)tbl;
    for (int i = tid; i < M_SUB * KSUB_ * BQ / 4; i += blockDim.x) d4[i] = s4[i];
    __syncthreads();
  }
#endif

  const int stride = gridDim.x * blockDim.x;
  for (int n = blockIdx.x * blockDim.x + tid; n < N; n += stride) {
    const uint4* cp4 = (const uint4*)(packed + (size_t)n * 16);
    __builtin_prefetch(packed + ((size_t)n + stride) * 16, 0, 0);
    float acc0 = 0.f, acc1 = 0.f, acc2 = 0.f, acc3 = 0.f;
#pragma unroll
    for (int g = 0; g < 4; ++g) {
      uint4 w = cp4[g];                    // 16 byte-codes per b128 load
      uint32_t words[4] = {w.x, w.y, w.z, w.w};
#pragma unroll
      for (int jw = 0; jw < 4; ++jw) {
        uint32_t cw4 = words[jw];
#pragma unroll
        for (int j = 0; j < 4; ++j) {
          int m = (g * 4 + jw) * 4 + j;
          int c = (cw4 >> (8 * j)) & 255;
          const float4 t = *(const float4*)(tbl + (((m << 8) + c) << 2));
          acc0 += t.x; acc1 += t.y; acc2 += t.z; acc3 += t.w;
        }
      }
    }
    dists[(size_t)(b0 + 0) * N + n] = acc0;
    dists[(size_t)(b0 + 1) * N + n] = acc1;
    dists[(size_t)(b0 + 2) * N + n] = acc2;
    dists[(size_t)(b0 + 3) * N + n] = acc3;
  }
}

// ---------------------------------------------------------------------------
// Kernel 4: per-(query, slice) exact top-64 via 64 rounds of block arg-min
// over an LDS-resident slice. Deterministic tie-break on smaller index.
// ---------------------------------------------------------------------------
__global__ __launch_bounds__(256)
void pq_select1(const float* __restrict__ dists,
                float* __restrict__ cand_d, int* __restrict__ cand_i,
                int N, int slice_len) {
  __shared__ float sd[SLICE_CAP];
  __shared__ float rmin[256];
  __shared__ int   ridx[256];
  const int slice = blockIdx.x, b = blockIdx.y, tid = threadIdx.x;
  const int start = slice * slice_len;
  const int len   = min(slice_len, N - start);
  const float* src = dists + (size_t)b * N + start;

  for (int i = tid; i < slice_len; i += blockDim.x)
    sd[i] = (i < len) ? src[i] : FLT_INF;
  __syncthreads();

  for (int it = 0; it < TOPK; ++it) {
    float mv = FLT_INF; int mi = 0x7fffffff;
    for (int i = tid; i < slice_len; i += blockDim.x) {
      float v = sd[i];
      if (v < mv || (v == mv && i < mi)) { mv = v; mi = i; }
    }
    rmin[tid] = mv; ridx[tid] = mi;
    __syncthreads();
    for (int s = 128; s > 0; s >>= 1) {
      if (tid < s) {
        if (rmin[tid + s] < rmin[tid] ||
            (rmin[tid + s] == rmin[tid] && ridx[tid + s] < ridx[tid])) {
          rmin[tid] = rmin[tid + s]; ridx[tid] = ridx[tid + s];
        }
      }
      __syncthreads();
    }
    if (tid == 0) {
      int o = (b * NSLICE + slice) * TOPK + it;
      cand_d[o] = rmin[0];
      cand_i[o] = start + ridx[0];
      sd[ridx[0]] = FLT_INF;               // knock out the winner
    }
    __syncthreads();
  }
}

// ---------------------------------------------------------------------------
// Kernel 5: merge 64*64 candidates per query into the final top-64.
// ---------------------------------------------------------------------------
__global__ __launch_bounds__(256)
void pq_select2(const float* __restrict__ cand_d,
                const int* __restrict__ cand_i,
                float* __restrict__ out_d, int* __restrict__ out_i) {
  __shared__ float sd[NSLICE * TOPK];
  __shared__ int   si[NSLICE * TOPK];
  __shared__ float rmin[256];
  __shared__ int   rgid[256];
  __shared__ int   rloc[256];
  const int b = blockIdx.x, tid = threadIdx.x;

  for (int i = tid; i < NSLICE * TOPK; i += blockDim.x) {
    sd[i] = cand_d[(size_t)b * NSLICE * TOPK + i];
    si[i] = cand_i[(size_t)b * NSLICE * TOPK + i];
  }
  __syncthreads();

  for (int it = 0; it < TOPK; ++it) {
    float mv = FLT_INF; int mg = 0x7fffffff; int ml = -1;
    for (int i = tid; i < NSLICE * TOPK; i += blockDim.x) {
      float v = sd[i]; int g = si[i];
      if (v < mv || (v == mv && g < mg)) { mv = v; mg = g; ml = i; }
    }
    rmin[tid] = mv; rgid[tid] = mg; rloc[tid] = ml;
    __syncthreads();
    for (int s = 128; s > 0; s >>= 1) {
      if (tid < s) {
        if (rmin[tid + s] < rmin[tid] ||
            (rmin[tid + s] == rmin[tid] && rgid[tid + s] < rgid[tid])) {
          rmin[tid] = rmin[tid + s];
          rgid[tid] = rgid[tid + s];
          rloc[tid] = rloc[tid + s];
        }
      }
      __syncthreads();
    }
    if (tid == 0) {
      out_d[b * TOPK + it] = rmin[0];
      out_i[b * TOPK + it] = rgid[0];
      if (rloc[0] >= 0) sd[rloc[0]] = FLT_INF;
    }
    __syncthreads();
  }
}

// ---------------------------------------------------------------------------
extern "C" void kernel_launch(void* const* d_in, const int* in_sizes, int n_in,
                              void* d_out, int out_size, void* d_ws, size_t ws_size,
                              hipStream_t stream) {
  const float* querys    = (const float*)d_in[0];   // (32, 256)
  const float* codewords = (const float*)d_in[1];   // (64, 256, 4)
  const int*   codes     = (const int*)d_in[2];     // (N, 64)
  const int N = in_sizes[2] / M_SUB;                // 1,000,000

  // Workspace layout
  char* ws = (char*)d_ws;
  size_t off = 0;
  float*    tableI = (float*)(ws + off); off += (size_t)BATCH * M_SUB * KSUB_ * 4; // 2MB
  uint32_t* packed = (uint32_t*)(ws + off); off += (size_t)N * M_SUB;              // 64MB
  off = (off + 255) & ~(size_t)255;
  float*    dists  = (float*)(ws + off); off += (size_t)BATCH * N * 4;             // 128MB
  float*    cand_d = (float*)(ws + off); off += (size_t)BATCH * NSLICE * TOPK * 4; // 0.5MB
  int*      cand_i = (int*)(ws + off);                                             // 0.5MB

  // 1) WMMA distance table (interleaved 4-query groups)
  pq_table_wmma<<<dim3(M_SUB, 2), 32, 0, stream>>>(querys, codewords, tableI);

  // 2) pack codes to u8
  pq_pack_codes<<<(N + 255) / 256, 256, 0, stream>>>(codes, packed, N);

  // 3) ADC scan, 4-query groups, TDM-staged 256KB LDS table per block
  const size_t scan_lds = (size_t)BQ * M_SUB * KSUB_ * sizeof(float);
  hipFuncSetAttribute((const void*)pq_scan,
                      hipFuncAttributeMaxDynamicSharedMemorySize, (int)scan_lds);
  pq_scan<<<dim3(256, BATCH / BQ), 256, scan_lds, stream>>>(packed, tableI, dists, N);

  // 4) slice-wise exact top-64
  const int slice_len = (N + NSLICE - 1) / NSLICE;   // 15625 <= SLICE_CAP
  pq_select1<<<dim3(NSLICE, BATCH), 256, 0, stream>>>(dists, cand_d, cand_i, N, slice_len);

  // 5) merge to final top-64; d_out = [dists (32x64 f32) | indices (32x64 i32)]
  pq_select2<<<BATCH, 256, 0, stream>>>(cand_d, cand_i,
                                        (float*)d_out,
                                        (int*)d_out + BATCH * TOPK);
}